// DeepGCN_70858370450154
// MI455X (gfx1250) — compile-verified
//
#include <hip/hip_runtime.h>

// ---------------------------------------------------------------------------
// DeepGCN / GENConv forward for MI455X (gfx1250, wave32).
// fp32 WMMA 16x16x4 for all GEMMs (workload is memory-bound; fp32 matches ref).
// ---------------------------------------------------------------------------

typedef float v2f __attribute__((ext_vector_type(2)));
typedef float v8f __attribute__((ext_vector_type(8)));

#define WMMA_F32(a, b, c) \
  __builtin_amdgcn_wmma_f32_16x16x4_f32(false, (a), false, (b), (short)0, (c), false, false)

constexpr int Nn  = 50000;
constexpr int Ee  = 500000;
constexpr int Cc  = 256;
constexpr int Hh  = 128;
constexpr int H2  = 256;   // 2*H
constexpr int Ll  = 4;
constexpr int Gg  = 64;
constexpr int Kk  = 8;
constexpr int NCc = 2;
constexpr float EPS_MSG = 1e-7f;
constexpr float EPS_BN  = 1e-5f;

// ---------------------------------------------------------------------------
// Kernel 1: fused input projections  xs = x@Wsrc+bsrc, xd = x@Wdst+bdst
// block = 256 threads (8 waves); block tile = 16 rows x 128 cols; K = 256.
// Each wave owns one 16x16 column tile and accumulates both projections,
// reusing the shared A tile (x rows) from LDS.
// ---------------------------------------------------------------------------
__global__ __launch_bounds__(256) void proj_kernel(
    const float* __restrict__ x,
    const float* __restrict__ wsrc, const float* __restrict__ bsrc,
    const float* __restrict__ wdst, const float* __restrict__ bdst,
    float* __restrict__ xs, float* __restrict__ xd)
{
  __shared__ float As[16 * Cc];                       // 16 KB
  const int row0 = blockIdx.x * 16;
  const int tid  = threadIdx.x;
  for (int idx = tid; idx < 16 * Cc; idx += 256)
    As[idx] = x[(size_t)(row0 + (idx >> 8)) * Cc + (idx & (Cc - 1))];
  __syncthreads();

  const int wave = tid >> 5, lane = tid & 31;
  const int hx = lane >> 4, r = lane & 15;            // lane half, lane row
  const int col = wave * 16 + r;

  v8f cs = {}, cd = {};
  for (int k0 = 0; k0 < Cc; k0 += 4) {
    const int ka = k0 + 2 * hx;
    v2f a, b0, b1;
    a.x  = As[r * Cc + ka];
    a.y  = As[r * Cc + ka + 1];
    b0.x = wsrc[(size_t)ka * Hh + col];
    b0.y = wsrc[(size_t)(ka + 1) * Hh + col];
    b1.x = wdst[(size_t)ka * Hh + col];
    b1.y = wdst[(size_t)(ka + 1) * Hh + col];
    cs = WMMA_F32(a, b0, cs);
    cd = WMMA_F32(a, b1, cd);
  }
  const float bs = bsrc[col], bd = bdst[col];
#pragma unroll
  for (int i = 0; i < 8; ++i) {
    const size_t gi = (size_t)(row0 + i + 8 * hx) * Hh + col;
    xs[gi] = cs[i] + bs;
    xd[gi] = cd[i] + bd;
  }
}

// ---------------------------------------------------------------------------
// Edge pass 1: z = (relu(xs[src] + edge_attr*ew + eb) + eps) * t ;
// atomic max of z into zmax[dst]. z > 0 always, so int-bitcast atomicMax on a
// zero-initialized buffer == segment_max with the isfinite->0 fixup.
// One wave per edge; lane handles 4 contiguous channels (coalesced float4).
// ---------------------------------------------------------------------------
__global__ __launch_bounds__(256) void edge_max_kernel(
    const int* __restrict__ ei, const float* __restrict__ ea,
    const float* __restrict__ xsrc,
    const float* __restrict__ ew, const float* __restrict__ ebv,
    const float* __restrict__ tp, float* __restrict__ zmax)
{
  const int e    = blockIdx.x * 8 + (threadIdx.x >> 5);
  const int lane = threadIdx.x & 31;
  const int src  = ei[e], dst = ei[Ee + e];
  const float av = ea[e], tl = tp[0];
  const int h = lane * 4;
  float xv[4];
  *(float4*)xv = *(const float4*)(xsrc + (size_t)src * Hh + h);
#pragma unroll
  for (int j = 0; j < 4; ++j) {
    const float m = fmaxf(xv[j] + av * ew[h + j] + ebv[h + j], 0.0f) + EPS_MSG;
    const float z = m * tl;
    atomicMax((int*)(zmax + (size_t)dst * Hh + h + j), __float_as_int(z));
  }
}

// ---------------------------------------------------------------------------
// Edge pass 2: ez = exp(z - zmax[dst]); den[dst] += ez; num[dst] += msg*ez.
// ---------------------------------------------------------------------------
__global__ __launch_bounds__(256) void edge_acc_kernel(
    const int* __restrict__ ei, const float* __restrict__ ea,
    const float* __restrict__ xsrc,
    const float* __restrict__ ew, const float* __restrict__ ebv,
    const float* __restrict__ tp, const float* __restrict__ zmax,
    float* __restrict__ den, float* __restrict__ num)
{
  const int e    = blockIdx.x * 8 + (threadIdx.x >> 5);
  const int lane = threadIdx.x & 31;
  const int src  = ei[e], dst = ei[Ee + e];
  const float av = ea[e], tl = tp[0];
  const int h = lane * 4;
  float xv[4];
  *(float4*)xv = *(const float4*)(xsrc + (size_t)src * Hh + h);
#pragma unroll
  for (int j = 0; j < 4; ++j) {
    const size_t gi = (size_t)dst * Hh + h + j;
    const float m  = fmaxf(xv[j] + av * ew[h + j] + ebv[h + j], 0.0f) + EPS_MSG;
    const float z  = m * tl;
    const float ez = expf(z - zmax[gi]);
    atomicAdd(den + gi, ez);
    atomicAdd(num + gi, m * ez);
  }
}

// ---------------------------------------------------------------------------
// Kernel: fused GENConv MLP for one 16-row stripe:
//   out  = num/max(den,1e-16) + skip          (computed into LDS A tile)
//   mid  = relu(BN(out @ w1 + b1))            (16x256, kept in LDS)
//   y    = mid @ w2 + b2
//   h    = y   (layer 0)   or   h += y   (residual layers)
// ---------------------------------------------------------------------------
__global__ __launch_bounds__(256) void genconv_mlp_kernel(
    const float* __restrict__ num, const float* __restrict__ den,
    const float* __restrict__ skip,
    const float* __restrict__ w1, const float* __restrict__ b1,
    const float* __restrict__ bng, const float* __restrict__ bnb,
    const float* __restrict__ bnm, const float* __restrict__ bnv,
    const float* __restrict__ w2, const float* __restrict__ b2,
    float* __restrict__ h, const int residual)
{
  __shared__ float As[16 * Hh];    // 8 KB : combined conv output (A of GEMM1)
  __shared__ float Mid[16 * H2];   // 16 KB: post-BN/ReLU intermediate
  const int row0 = blockIdx.x * 16;
  const int tid  = threadIdx.x;

  for (int idx = tid; idx < 16 * Hh; idx += 256) {
    const size_t gi = (size_t)(row0 + (idx >> 7)) * Hh + (idx & (Hh - 1));
    As[idx] = num[gi] / fmaxf(den[gi], 1e-16f) + skip[gi];
  }
  __syncthreads();

  const int wave = tid >> 5, lane = tid & 31;
  const int hx = lane >> 4, r = lane & 15;

  // Phase 1: mid[16x256] = relu(BN(As[16x128] @ w1[128x256] + b1))
#pragma unroll
  for (int tt = 0; tt < 2; ++tt) {
    const int colw = wave * 32 + tt * 16;
    const int col  = colw + r;
    v8f c = {};
    for (int k0 = 0; k0 < Hh; k0 += 4) {
      const int ka = k0 + 2 * hx;
      v2f a, b;
      a.x = As[r * Hh + ka];
      a.y = As[r * Hh + ka + 1];
      b.x = w1[(size_t)ka * H2 + col];
      b.y = w1[(size_t)(ka + 1) * H2 + col];
      c = WMMA_F32(a, b, c);
    }
    const float sc = bng[col] * rsqrtf(bnv[col] + EPS_BN);
    const float mB = bnm[col], bB = bnb[col], bi = b1[col];
#pragma unroll
    for (int i = 0; i < 8; ++i) {
      const float val = sc * ((c[i] + bi) - mB) + bB;
      Mid[(i + 8 * hx) * H2 + col] = fmaxf(val, 0.0f);
    }
  }
  __syncthreads();

  // Phase 2: y[16x128] = Mid[16x256] @ w2[256x128] + b2
  {
    const int col = wave * 16 + r;
    v8f c = {};
    for (int k0 = 0; k0 < H2; k0 += 4) {
      const int ka = k0 + 2 * hx;
      v2f a, b;
      a.x = Mid[r * H2 + ka];
      a.y = Mid[r * H2 + ka + 1];
      b.x = w2[(size_t)ka * Hh + col];
      b.y = w2[(size_t)(ka + 1) * Hh + col];
      c = WMMA_F32(a, b, c);
    }
    const float bi = b2[col];
#pragma unroll
    for (int i = 0; i < 8; ++i) {
      const size_t gi = (size_t)(row0 + i + 8 * hx) * Hh + col;
      const float val = c[i] + bi;
      h[gi] = residual ? (h[gi] + val) : val;
    }
  }
}

// ---------------------------------------------------------------------------
// Elementwise r = relu(BN(h)) with running stats.
// ---------------------------------------------------------------------------
__global__ __launch_bounds__(256) void bnrelu_kernel(
    const float* __restrict__ hin,
    const float* __restrict__ g, const float* __restrict__ b,
    const float* __restrict__ m, const float* __restrict__ v,
    float* __restrict__ out)
{
  const size_t i = (size_t)blockIdx.x * 256 + threadIdx.x;
  const int hc = (int)(i & (Hh - 1));
  const float sc = g[hc] * rsqrtf(v[hc] + EPS_BN);
  out[i] = fmaxf(sc * (hin[i] - m[hc]) + b[hc], 0.0f);
}

// ---------------------------------------------------------------------------
// Global mean pool (accumulate phase): one block per node, 128 threads.
// ---------------------------------------------------------------------------
__global__ __launch_bounds__(128) void pool_kernel(
    const float* __restrict__ hf, const int* __restrict__ batch,
    float* __restrict__ pooled, float* __restrict__ cnt)
{
  const int i = blockIdx.x;
  const int b = batch[i];
  atomicAdd(pooled + (size_t)b * Hh + threadIdx.x,
            hf[(size_t)i * Hh + threadIdx.x]);
  if (threadIdx.x == 0) atomicAdd(cnt + b, 1.0f);
}

// ---------------------------------------------------------------------------
// Classifier: out[g,c] = concat(pooled_mean[g], clinical[g]) @ cls_w + cls_b.
// G*NC = 128 outputs -> one block.
// ---------------------------------------------------------------------------
__global__ __launch_bounds__(128) void classifier_kernel(
    const float* __restrict__ pooled, const float* __restrict__ cnt,
    const float* __restrict__ clin,
    const float* __restrict__ w, const float* __restrict__ b,
    float* __restrict__ out)
{
  const int t = threadIdx.x;
  const int g = t >> 1, c = t & 1;
  const float inv = 1.0f / fmaxf(cnt[g], 1.0f);
  float acc = b[c];
  for (int hh = 0; hh < Hh; ++hh)
    acc += pooled[(size_t)g * Hh + hh] * inv * w[hh * NCc + c];
  for (int k = 0; k < Kk; ++k)
    acc += clin[g * Kk + k] * w[(Hh + k) * NCc + c];
  out[t] = acc;
}

// ---------------------------------------------------------------------------
// Host orchestration (graph-capture safe: only kernel launches + MemsetAsync).
// ---------------------------------------------------------------------------
extern "C" void kernel_launch(void* const* d_in, const int* in_sizes, int n_in,
                              void* d_out, int out_size, void* d_ws, size_t ws_size,
                              hipStream_t stream) {
  const float* x    = (const float*)d_in[0];
  const int*   ei   = (const int*)  d_in[1];
  const float* ea   = (const float*)d_in[2];
  const int*   bat  = (const int*)  d_in[3];
  const float* clin = (const float*)d_in[4];
  const float* lsw  = (const float*)d_in[5];
  const float* lsb  = (const float*)d_in[6];
  const float* ldw  = (const float*)d_in[7];
  const float* ldb2 = (const float*)d_in[8];
  const float* ew   = (const float*)d_in[9];
  const float* ebv  = (const float*)d_in[10];
  const float* t    = (const float*)d_in[11];
  const float* w1   = (const float*)d_in[12];
  const float* b1   = (const float*)d_in[13];
  const float* bng  = (const float*)d_in[14];
  const float* bnb  = (const float*)d_in[15];
  const float* bnm  = (const float*)d_in[16];
  const float* bnv  = (const float*)d_in[17];
  const float* w2   = (const float*)d_in[18];
  const float* b2   = (const float*)d_in[19];
  const float* ng   = (const float*)d_in[20];
  const float* nb   = (const float*)d_in[21];
  const float* nmr  = (const float*)d_in[22];
  const float* nvr  = (const float*)d_in[23];
  const float* cw   = (const float*)d_in[24];
  const float* cb   = (const float*)d_in[25];

  const size_t NH = (size_t)Nn * Hh;
  float* ws0    = (float*)d_ws;
  float* xs     = ws0 + 0 * NH;  // layer-0 src feats; recycled as r and final act
  float* xd     = ws0 + 1 * NH;  // layer-0 dst skip feats
  float* h      = ws0 + 2 * NH;
  float* zm     = ws0 + 3 * NH;  // zmax | den | num are contiguous (one memset)
  float* den    = ws0 + 4 * NH;
  float* numb   = ws0 + 5 * NH;
  float* pooled = ws0 + 6 * NH;  // G*H
  float* cnt    = pooled + (size_t)Gg * Hh;

  // Input projections (layer 0 feeds)
  proj_kernel<<<Nn / 16, 256, 0, stream>>>(x, lsw, lsb, ldw, ldb2, xs, xd);

  for (int l = 0; l < Ll; ++l) {
    if (l > 0) {
      // r = relu(BN_l(h)) -> xs slot; serves as both msg source and skip.
      bnrelu_kernel<<<NH / 256, 256, 0, stream>>>(
          h, ng + l * Hh, nb + l * Hh, nmr + l * Hh, nvr + l * Hh, xs);
    }
    hipMemsetAsync(zm, 0, 3 * NH * sizeof(float), stream);  // zmax, den, num
    edge_max_kernel<<<Ee / 8, 256, 0, stream>>>(
        ei, ea, xs, ew + l * Hh, ebv + l * Hh, t + l, zm);
    edge_acc_kernel<<<Ee / 8, 256, 0, stream>>>(
        ei, ea, xs, ew + l * Hh, ebv + l * Hh, t + l, zm, den, numb);
    const float* skip = (l == 0) ? xd : xs;
    genconv_mlp_kernel<<<Nn / 16, 256, 0, stream>>>(
        numb, den, skip,
        w1 + (size_t)l * Hh * H2, b1 + l * H2,
        bng + l * H2, bnb + l * H2, bnm + l * H2, bnv + l * H2,
        w2 + (size_t)l * H2 * Hh, b2 + l * Hh,
        h, (l > 0) ? 1 : 0);
  }

  // Final norm/act (layer-0 params), pool, classify.
  bnrelu_kernel<<<NH / 256, 256, 0, stream>>>(h, ng, nb, nmr, nvr, xs);
  hipMemsetAsync(pooled, 0, ((size_t)Gg * Hh + Gg) * sizeof(float), stream);
  pool_kernel<<<Nn, 128, 0, stream>>>(xs, bat, pooled, cnt);
  classifier_kernel<<<1, Gg * NCc, 0, stream>>>(pooled, cnt, clin, cw, cb,
                                                (float*)d_out);
}